// Memory_38482906972752
// MI455X (gfx1250) — compile-verified
//
#include <hip/hip_runtime.h>
#include <hip/hip_bf16.h>
#include <math.h>

#define BB 16
#define SS 2048
#define DD 64
#define HH 4
#define EPSF 1e-8f

typedef float v2f __attribute__((ext_vector_type(2)));
typedef float v8f __attribute__((ext_vector_type(8)));

#if defined(__HIP_DEVICE_COMPILE__)
#if !__has_builtin(__builtin_amdgcn_wmma_f32_16x16x4_f32)
#error "wmma_f32_16x16x4_f32 builtin not available on this toolchain"
#endif
#endif

__device__ __forceinline__ v8f wmma_f32_4(v2f a, v2f b, v8f c) {
  // D = A(16x4 f32) * B(4x16 f32) + C(16x16 f32)
  return __builtin_amdgcn_wmma_f32_16x16x4_f32(
      /*neg_a=*/false, a, /*neg_b=*/false, b,
      /*c_mod=*/(short)0, c, /*reuse_a=*/false, /*reuse_b=*/false);
}

// ---------------------------------------------------------------------------
// Kernel 1: usage update + write-key content lookup (softmax over S)
// ---------------------------------------------------------------------------
__global__ void k_prologue(const float* __restrict__ mem,
                           const float* __restrict__ usage,
                           const float* __restrict__ prev_w,
                           const float* __restrict__ prev_r,
                           const float* __restrict__ wkey,
                           const float* __restrict__ wstr,
                           const float* __restrict__ fgate,
                           float* __restrict__ ws_usage,
                           float* __restrict__ ws_lookup) {
  const int b = blockIdx.x;
  const int tid = threadIdx.x; // 256 threads
  __shared__ float kL[DD];
  __shared__ float red[256];
  if (tid < DD) kL[tid] = wkey[b * DD + tid];
  __syncthreads();

  float kn2 = 0.f;
#pragma unroll
  for (int d = 0; d < DD; ++d) kn2 += kL[d] * kL[d];
  const float keynorm = sqrtf(kn2);
  const float strength = wstr[b];
  float fg[HH];
#pragma unroll
  for (int h = 0; h < HH; ++h) fg[h] = fgate[b * HH + h];

  float logit[8];
#pragma unroll
  for (int k = 0; k < 8; ++k) {
    const int s = tid + k * 256;
    const size_t rs = (size_t)b * SS + s;
    // usage update: (u + pw - u*pw) * prod_h (2 - r*f)
    const float u = usage[rs];
    const float pw = prev_w[rs];
    float ret = 1.f;
#pragma unroll
    for (int h = 0; h < HH; ++h) ret *= (2.f - prev_r[rs * HH + h] * fg[h]);
    ws_usage[rs] = (u + pw - u * pw) * ret;
    // cosine lookup logit
    const float4* mrow = (const float4*)(mem + rs * DD);
    float dot = 0.f, nrm = 0.f;
#pragma unroll
    for (int q = 0; q < DD / 4; ++q) {
      float4 m = mrow[q];
      dot += m.x * kL[4 * q + 0] + m.y * kL[4 * q + 1] + m.z * kL[4 * q + 2] + m.w * kL[4 * q + 3];
      nrm += m.x * m.x + m.y * m.y + m.z * m.z + m.w * m.w;
    }
    const float cosv = dot / (sqrtf(nrm) * keynorm + EPSF);
    logit[k] = cosv * strength;
  }
  // softmax over S
  float mx = logit[0];
#pragma unroll
  for (int k = 1; k < 8; ++k) mx = fmaxf(mx, logit[k]);
  red[tid] = mx; __syncthreads();
  for (int o = 128; o > 0; o >>= 1) { if (tid < o) red[tid] = fmaxf(red[tid], red[tid + o]); __syncthreads(); }
  mx = red[0]; __syncthreads();
  float ssum = 0.f;
#pragma unroll
  for (int k = 0; k < 8; ++k) ssum += expf(logit[k] - mx);
  red[tid] = ssum; __syncthreads();
  for (int o = 128; o > 0; o >>= 1) { if (tid < o) red[tid] += red[tid + o]; __syncthreads(); }
  ssum = red[0]; __syncthreads();
  const float inv = 1.f / ssum;
#pragma unroll
  for (int k = 0; k < 8; ++k) {
    const int s = tid + k * 256;
    ws_lookup[(size_t)b * SS + s] = expf(logit[k] - mx) * inv;
  }
}

// ---------------------------------------------------------------------------
// Kernel 2: allocation weights (bitonic sort + cumprod scan + scatter) and w
// ---------------------------------------------------------------------------
__global__ void k_alloc(const float* __restrict__ ws_usage,
                        const float* __restrict__ ws_lookup,
                        const float* __restrict__ wgate,
                        const float* __restrict__ agate,
                        float* __restrict__ ws_w) {
  const int b = blockIdx.x;
  const int tid = threadIdx.x; // 1024 threads
  __shared__ float val[SS];
  __shared__ int   idx[SS];
  __shared__ float scA[SS];
  __shared__ float scB[SS];

#pragma unroll
  for (int e = 0; e < 2; ++e) {
    const int i = tid + e * 1024;
    val[i] = ws_usage[(size_t)b * SS + i];
    idx[i] = i;
  }
  __syncthreads();

  // bitonic sort ascending by value (index payload)
  for (int k2 = 2; k2 <= SS; k2 <<= 1) {
    for (int j = k2 >> 1; j > 0; j >>= 1) {
#pragma unroll
      for (int e = 0; e < 2; ++e) {
        const int i = tid + e * 1024;
        const int p = i ^ j;
        if (p > i) {
          const bool up = ((i & k2) == 0);
          const float vi = val[i], vp = val[p];
          if ((vi > vp) == up) {
            val[i] = vp; val[p] = vi;
            const int t = idx[i]; idx[i] = idx[p]; idx[p] = t;
          }
        }
      }
      __syncthreads();
    }
  }

  // inclusive cumprod (Hillis-Steele, ping-pong)
#pragma unroll
  for (int e = 0; e < 2; ++e) { const int i = tid + e * 1024; scA[i] = val[i]; }
  __syncthreads();
  float* sa = scA; float* sb = scB;
  for (int off = 1; off < SS; off <<= 1) {
#pragma unroll
    for (int e = 0; e < 2; ++e) {
      const int i = tid + e * 1024;
      float x = sa[i];
      if (i >= off) x *= sa[i - off];
      sb[i] = x;
    }
    __syncthreads();
    float* t = sa; sa = sb; sb = t;
  }

  const float wg = wgate[b], ag = agate[b];
#pragma unroll
  for (int e = 0; e < 2; ++e) {
    const int i = tid + e * 1024;
    const float ex = (i == 0) ? 1.f : sa[i - 1];       // exclusive cumprod
    const float alloc = (1.f - val[i]) * ex;            // unordered alloc
    const int s = idx[i];                               // scatter to free list
    const float lw = ws_lookup[(size_t)b * SS + s];
    ws_w[(size_t)b * SS + s] = wg * (ag * alloc + (1.f - ag) * lw);
  }
}

// ---------------------------------------------------------------------------
// Kernel 3: stream link, fuse link_new update + fwd/bwd einsums via WMMA f32
//   grid: (B, S/128 row bands, S/256 col bands), 256 threads = 8 waves
//   wave owns 16 rows; fwd accumulator lives in v8f across the column band;
//   bwd partials accumulate in LDS, one atomic flush per block.
//   rT staged zero-padded to 16 N-rows (stride +2 for bank spread) so the
//   WMMA B operand is a plain b64 LDS load, no masking VALU in the hot loop.
// ---------------------------------------------------------------------------
#define ROWS_PB 128
#define COLS_PB 256
#define LT_STRIDE 18        // 16 + 2 pad: 8B-aligned b64 rows, spread banks
#define RTC_STRIDE (COLS_PB + 2)
#define RTR_STRIDE (ROWS_PB + 2)

__global__ void k_link(const float* __restrict__ link,
                       const float* __restrict__ prec,
                       const float* __restrict__ prev_r,
                       const float* __restrict__ ws_w,
                       float* __restrict__ ws_fwd,
                       float* __restrict__ ws_bwd) {
  const int b = blockIdx.x;
  const int i_base = blockIdx.y * ROWS_PB;
  const int j_base = blockIdx.z * COLS_PB;
  const int tid = threadIdx.x;
  const int wv = tid >> 5;
  const int lane = tid & 31;
  const int hN = lane & 15;     // N (output column / h) index
  const int halfk = lane >> 4;  // lane half selects K offset

  __shared__ float w_row[ROWS_PB];
  __shared__ float w_col[COLS_PB];
  __shared__ float prec_col[COLS_PB];
  __shared__ float rT_col[16 * RTC_STRIDE];  // prev_read^T, rows 4..15 zero
  __shared__ float rT_row[16 * RTR_STRIDE];  // prev_read^T, rows 4..15 zero
  __shared__ float bwd_part[COLS_PB * HH];
  __shared__ float ltile[8][16 * LT_STRIDE];

  if (tid < ROWS_PB) w_row[tid] = ws_w[(size_t)b * SS + i_base + tid];
  {
    w_col[tid] = ws_w[(size_t)b * SS + j_base + tid];
    prec_col[tid] = prec[(size_t)b * SS + j_base + tid];
  }
  for (int q = tid; q < 16 * COLS_PB; q += 256) {
    const int h = q >> 8, j = q & (COLS_PB - 1);
    rT_col[h * RTC_STRIDE + j] =
        (h < HH) ? prev_r[((size_t)b * SS + j_base + j) * HH + h] : 0.f;
  }
  for (int q = tid; q < 16 * ROWS_PB; q += 256) {
    const int h = q >> 7, j = q & (ROWS_PB - 1);
    rT_row[h * RTR_STRIDE + j] =
        (h < HH) ? prev_r[((size_t)b * SS + i_base + j) * HH + h] : 0.f;
  }
  for (int q = tid; q < COLS_PB * HH; q += 256) bwd_part[q] = 0.f;
  __syncthreads();

  const int il = lane >> 1;          // local row this lane loads (0..15)
  const int cs = (lane & 1) * 8;     // column sub-offset (0 or 8)
  const int gi = i_base + wv * 16 + il;
  const float wi = w_row[wv * 16 + il];
  float* lt = &ltile[wv][0];
  // row band intersects column band -> diagonal may appear (1/8 of blocks)
  const bool mayDiag = (i_base < j_base + COLS_PB) && (j_base < i_base + ROWS_PB);

  v8f accF = {0.f, 0.f, 0.f, 0.f, 0.f, 0.f, 0.f, 0.f};

  for (int jt = 0; jt < COLS_PB / 16; ++jt) {
    // ---- build link_new 16x16 tile in LDS (per wave) ----
    const size_t rowoff = ((size_t)b * SS + gi) * SS + j_base + jt * 16 + cs;
    const float4 x0 = *(const float4*)(link + rowoff);
    const float4 x1 = *(const float4*)(link + rowoff + 4);
    if (jt + 1 < COLS_PB / 16)
      __builtin_prefetch(link + rowoff + 16, 0, 0);  // global_prefetch_b8
    float xs[8];
    *(float4*)&xs[0] = x0; *(float4*)&xs[4] = x1;
#pragma unroll
    for (int e = 0; e < 8; ++e) {
      const int jl = jt * 16 + cs + e;               // local col in band
      lt[il * LT_STRIDE + cs + e] =
          (1.f - wi - w_col[jl]) * xs[e] + wi * prec_col[jl];
    }
    if (mayDiag) {                                   // zero link_new diagonal
      const int dj = gi - (j_base + jt * 16 + cs);
      if (dj >= 0 && dj < 8) lt[il * LT_STRIDE + cs + dj] = 0.f;
    }
    __builtin_amdgcn_wave_barrier();  // per-wave tile: order LDS write->read

    // ---- fwd: accF(16 rows x 16 cols[h-pad]) += Ltile * r ----
#pragma unroll
    for (int c = 0; c < 4; ++c) {
      const int kk = c * 4 + 2 * halfk;              // K = local column
      v2f a; a.x = lt[hN * LT_STRIDE + kk];          // A[M=hN][K=kk..kk+1]
      a.y = lt[hN * LT_STRIDE + kk + 1];
      const int jl = jt * 16 + kk;
      v2f bb;                                        // B[K][N=h], rows>=4 zero
      bb.x = rT_col[hN * RTC_STRIDE + jl];
      bb.y = rT_col[hN * RTC_STRIDE + jl + 1];
      accF = wmma_f32_4(a, bb, accF);
    }

    // ---- bwd: tile^T * r -> 16 j-rows x h, flush per tile into LDS ----
    v8f accB = {0.f, 0.f, 0.f, 0.f, 0.f, 0.f, 0.f, 0.f};
#pragma unroll
    for (int c = 0; c < 4; ++c) {
      const int ik = c * 4 + 2 * halfk;              // K = local row
      v2f a; a.x = lt[ik * LT_STRIDE + hN];          // A[M=j=hN][K=i]
      a.y = lt[(ik + 1) * LT_STRIDE + hN];
      const int ir = wv * 16 + ik;
      v2f bb;
      bb.x = rT_row[hN * RTR_STRIDE + ir];
      bb.y = rT_row[hN * RTR_STRIDE + ir + 1];
      accB = wmma_f32_4(a, bb, accB);
    }
    if (hN < HH) {
#pragma unroll
      for (int v = 0; v < 8; ++v) {
        const int jl = jt * 16 + v + 8 * halfk;      // D: M = v + 8*(lane/16)
        atomicAdd(&bwd_part[jl * HH + hN], accB[v]); // ds_add_f32
      }
    }
  }

  // flush fwd accumulator (valid h lanes only)
  if (hN < HH) {
#pragma unroll
    for (int v = 0; v < 8; ++v) {
      const int gi2 = i_base + wv * 16 + v + 8 * halfk;
      atomicAdd(&ws_fwd[((size_t)b * SS + gi2) * HH + hN], accF[v]);
    }
  }
  __syncthreads();
  for (int q = tid; q < COLS_PB * HH; q += 256) {
    const int jl = q >> 2, h = q & 3;
    atomicAdd(&ws_bwd[((size_t)b * SS + j_base + jl) * HH + h], bwd_part[q]);
  }
}

// ---------------------------------------------------------------------------
// Kernel 4: content addressing of read keys against mem_new (recomputed)
// ---------------------------------------------------------------------------
__global__ void k_content(const float* __restrict__ mem,
                          const float* __restrict__ rkeys,
                          const float* __restrict__ rstr,
                          const float* __restrict__ ws_w,
                          const float* __restrict__ evec,
                          const float* __restrict__ wvec,
                          float* __restrict__ ws_content) {
  const int b = blockIdx.x;
  const int tid = threadIdx.x; // 256
  __shared__ float kL[DD * HH];
  __shared__ float eL[DD], vL[DD];
  __shared__ float red[256];
  if (tid < DD) { eL[tid] = evec[b * DD + tid]; vL[tid] = wvec[b * DD + tid]; }
  kL[tid] = rkeys[b * DD * HH + tid];
  __syncthreads();

  float kn[HH];
#pragma unroll
  for (int h = 0; h < HH; ++h) {
    float s2 = 0.f;
#pragma unroll
    for (int d = 0; d < DD; ++d) s2 += kL[d * HH + h] * kL[d * HH + h];
    kn[h] = sqrtf(s2);
  }
  float stg[HH];
#pragma unroll
  for (int h = 0; h < HH; ++h) stg[h] = rstr[b * HH + h];

  float lg[8][HH];
#pragma unroll
  for (int k = 0; k < 8; ++k) {
    const int s = tid + k * 256;
    const size_t rs = (size_t)b * SS + s;
    const float ww = ws_w[rs];
    float nrm = 0.f, dt[HH] = {0.f, 0.f, 0.f, 0.f};
#pragma unroll
    for (int d = 0; d < DD; ++d) {
      const float m = mem[rs * DD + d];
      const float mn = m * (1.f - ww * eL[d]) + ww * vL[d];  // mem_new on the fly
      nrm += mn * mn;
#pragma unroll
      for (int h = 0; h < HH; ++h) dt[h] += mn * kL[d * HH + h];
    }
    nrm = sqrtf(nrm);
#pragma unroll
    for (int h = 0; h < HH; ++h)
      lg[k][h] = dt[h] / (nrm * kn[h] + EPSF) * stg[h];
  }
  // 4 softmaxes over S
  for (int h = 0; h < HH; ++h) {
    float mx = lg[0][h];
#pragma unroll
    for (int k = 1; k < 8; ++k) mx = fmaxf(mx, lg[k][h]);
    red[tid] = mx; __syncthreads();
    for (int o = 128; o > 0; o >>= 1) { if (tid < o) red[tid] = fmaxf(red[tid], red[tid + o]); __syncthreads(); }
    mx = red[0]; __syncthreads();
    float sm = 0.f;
#pragma unroll
    for (int k = 0; k < 8; ++k) sm += expf(lg[k][h] - mx);
    red[tid] = sm; __syncthreads();
    for (int o = 128; o > 0; o >>= 1) { if (tid < o) red[tid] += red[tid + o]; __syncthreads(); }
    sm = red[0]; __syncthreads();
    const float inv = 1.f / sm;
#pragma unroll
    for (int k = 0; k < 8; ++k) {
      const int s = tid + k * 256;
      ws_content[((size_t)b * SS + s) * HH + h] = expf(lg[k][h] - mx) * inv;
    }
  }
}

// ---------------------------------------------------------------------------
// Kernel 5: read_w interpolation + read_vecs = mem_new^T @ read_w
//   grid (B, 8 s-chunks); out accumulated with atomics (out pre-zeroed)
// ---------------------------------------------------------------------------
__global__ void k_read(const float* __restrict__ mem,
                       const float* __restrict__ ws_w,
                       const float* __restrict__ evec,
                       const float* __restrict__ wvec,
                       const float* __restrict__ modes,
                       const float* __restrict__ ws_bwd,
                       const float* __restrict__ ws_content,
                       const float* __restrict__ ws_fwd,
                       float* __restrict__ out) {
  const int b = blockIdx.x;
  const int s0 = blockIdx.y * 256;
  const int tid = threadIdx.x; // 256
  const int d = tid & 63;
  const int h = tid >> 6;
  __shared__ float eL[DD], vL[DD];
  if (tid < DD) { eL[tid] = evec[b * DD + tid]; vL[tid] = wvec[b * DD + tid]; }
  __syncthreads();

  // softmax over the 3 read modes for this h
  const float m0 = modes[b * 12 + 0 * HH + h];
  const float m1 = modes[b * 12 + 1 * HH + h];
  const float m2 = modes[b * 12 + 2 * HH + h];
  const float mm = fmaxf(m0, fmaxf(m1, m2));
  const float e0 = expf(m0 - mm), e1 = expf(m1 - mm), e2 = expf(m2 - mm);
  const float pinv = 1.f / (e0 + e1 + e2);
  const float pi0 = e0 * pinv, pi1 = e1 * pinv, pi2 = e2 * pinv;

  const float ed = eL[d], vd = vL[d];
  float acc = 0.f;
  for (int s = s0; s < s0 + 256; ++s) {
    const size_t rs = (size_t)b * SS + s;
    const float ww = ws_w[rs];
    const float rw = pi0 * ws_bwd[rs * HH + h] + pi1 * ws_content[rs * HH + h]
                   + pi2 * ws_fwd[rs * HH + h];
    const float m = mem[rs * DD + d];
    const float mn = m * (1.f - ww * ed) + ww * vd;
    acc += mn * rw;
  }
  atomicAdd(&out[(b * DD + d) * HH + h], acc);
}

// ---------------------------------------------------------------------------
extern "C" void kernel_launch(void* const* d_in, const int* in_sizes, int n_in,
                              void* d_out, int out_size, void* d_ws, size_t ws_size,
                              hipStream_t stream) {
  (void)in_sizes; (void)n_in; (void)ws_size;
  const float* mem    = (const float*)d_in[0];
  const float* usage  = (const float*)d_in[1];
  const float* prec   = (const float*)d_in[2];
  const float* link   = (const float*)d_in[3];
  const float* prev_w = (const float*)d_in[4];
  const float* prev_r = (const float*)d_in[5];
  const float* wkey   = (const float*)d_in[6];
  const float* wstr   = (const float*)d_in[7];
  const float* fgate  = (const float*)d_in[8];
  const float* rkeys  = (const float*)d_in[9];
  const float* rstr   = (const float*)d_in[10];
  const float* wgate  = (const float*)d_in[11];
  const float* agate  = (const float*)d_in[12];
  const float* wvec   = (const float*)d_in[13];
  const float* evec   = (const float*)d_in[14];
  const float* modes  = (const float*)d_in[15];

  float* wsf = (float*)d_ws;
  float* ws_lookup  = wsf;                       // B*S
  float* ws_usage   = wsf + BB * SS;             // B*S
  float* ws_w       = wsf + 2 * BB * SS;         // B*S
  float* ws_fwd     = wsf + 3 * BB * SS;         // B*S*H
  float* ws_bwd     = ws_fwd + BB * SS * HH;     // B*S*H
  float* ws_content = ws_bwd + BB * SS * HH;     // B*S*H

  (void)hipMemsetAsync(ws_fwd, 0, (size_t)2 * BB * SS * HH * sizeof(float), stream);
  (void)hipMemsetAsync(d_out, 0, (size_t)out_size * sizeof(float), stream);

  k_prologue<<<BB, 256, 0, stream>>>(mem, usage, prev_w, prev_r, wkey, wstr,
                                     fgate, ws_usage, ws_lookup);
  k_alloc<<<BB, 1024, 0, stream>>>(ws_usage, ws_lookup, wgate, agate, ws_w);
  k_link<<<dim3(BB, SS / ROWS_PB, SS / COLS_PB), 256, 0, stream>>>(
      link, prec, prev_r, ws_w, ws_fwd, ws_bwd);
  k_content<<<BB, 256, 0, stream>>>(mem, rkeys, rstr, ws_w, evec, wvec,
                                    ws_content);
  k_read<<<dim3(BB, SS / 256), 256, 0, stream>>>(mem, ws_w, evec, wvec, modes,
                                                 ws_bwd, ws_content, ws_fwd,
                                                 (float*)d_out);
}